// CAFVBlock_90769838834068
// MI455X (gfx1250) — compile-verified
//
#include <hip/hip_runtime.h>
#include <math.h>

typedef __attribute__((ext_vector_type(2))) float v2f;
typedef __attribute__((ext_vector_type(8))) float v8f;

#define B_  4
#define Ca_ 128
#define Cv_ 512
#define NH_ 8
#define Ta_ 64
#define F_  64
#define Tv_ 256

// workspace layout (float offsets)
#define WS_ROWSUM 0                            // B*Ca*Ta = 32768
#define WS_S1     (WS_ROWSUM + B_*Ca_*Ta_)     // B*Ca = 512
#define WS_S2     (WS_S1 + B_*Ca_)             // 512
#define WS_SV1    (WS_S2 + B_*Ca_)             // B*Cv = 2048
#define WS_SV2    (WS_SV1 + B_*Cv_)            // 2048
#define WS_SC     (WS_SV2 + B_*Cv_)            // B*8 = 32

__device__ __forceinline__ float blockReduceSum(float v) {
    __shared__ float tmp[8];
    for (int o = 16; o > 0; o >>= 1) v += __shfl_xor(v, o);
    int w = threadIdx.x >> 5;
    if ((threadIdx.x & 31) == 0) tmp[w] = v;
    __syncthreads();
    float s = 0.f;
    int nw = blockDim.x >> 5;
    for (int i = 0; i < nw; ++i) s += tmp[i];
    __syncthreads();
    return s;
}

__device__ __forceinline__ float blockReduceMax(float v) {
    __shared__ float tmp[8];
    for (int o = 16; o > 0; o >>= 1) v = fmaxf(v, __shfl_xor(v, o));
    int w = threadIdx.x >> 5;
    if ((threadIdx.x & 31) == 0) tmp[w] = v;
    __syncthreads();
    float r = tmp[0];
    int nw = blockDim.x >> 5;
    for (int i = 1; i < nw; ++i) r = fmaxf(r, tmp[i]);
    __syncthreads();
    return r;
}

// K1: per-channel stats. Blocks [0, B*Ca): audio rowsums + S1/S2.
//     Blocks [B*Ca, B*Ca + B*Cv): video Sv1/Sv2.
__global__ __launch_bounds__(256) void k_stats(const float* __restrict__ audio,
                                               const float* __restrict__ video,
                                               float* __restrict__ ws) {
    int blk = blockIdx.x;
    int tid = threadIdx.x;
    if (blk < B_ * Ca_) {
        const float* ab = audio + (size_t)blk * (Ta_ * F_);
        int row = tid >> 2, k = tid & 3;           // 4 threads per 64-wide row
        const float4* p = (const float4*)(ab + row * F_ + k * 16);
        float s = 0.f, sq = 0.f;
        for (int i = 0; i < 4; ++i) {
            float4 x = p[i];
            s  += x.x + x.y + x.z + x.w;
            sq += x.x * x.x + x.y * x.y + x.z * x.z + x.w * x.w;
        }
        float rsum = s;
        rsum += __shfl_xor(rsum, 1);               // combine the 4 partials/row
        rsum += __shfl_xor(rsum, 2);
        if (k == 0) ws[WS_ROWSUM + blk * Ta_ + row] = rsum;
        float S1 = blockReduceSum(s);
        float S2 = blockReduceSum(sq);
        if (tid == 0) { ws[WS_S1 + blk] = S1; ws[WS_S2 + blk] = S2; }
    } else {
        int bc = blk - B_ * Ca_;                   // (b*Cv + c)
        float x = video[(size_t)bc * Tv_ + tid];
        float S1 = blockReduceSum(x);
        float S2 = blockReduceSum(x * x);
        if (tid == 0) { ws[WS_SV1 + bc] = S1; ws[WS_SV2 + bc] = S2; }
    }
}

// K2: one block per batch; fold per-channel stats with weights -> mu/rs per norm.
__global__ __launch_bounds__(256) void k_scalars(const float* __restrict__ p1_w, const float* __restrict__ p1_b,
                                                 const float* __restrict__ p2_w, const float* __restrict__ p2_b,
                                                 const float* __restrict__ f1_w, const float* __restrict__ f1_b,
                                                 const float* __restrict__ f2_w, const float* __restrict__ f2_b,
                                                 float* __restrict__ ws) {
    int b = blockIdx.x;
    int tid = threadIdx.x;
    const float* S1  = ws + WS_S1  + b * Ca_;
    const float* S2  = ws + WS_S2  + b * Ca_;
    const float* Sv1 = ws + WS_SV1 + b * Cv_;
    const float* Sv2 = ws + WS_SV2 + b * Cv_;

    float a_sw = 0, a_sww = 0, a_swb = 0, a_sb = 0, a_sbb = 0;
    float g_sw = 0, g_sww = 0, g_swb = 0, g_sb = 0, g_sbb = 0;
    for (int c = tid; c < Cv_; c += blockDim.x) {
        float s1 = S1[c >> 2], s2 = S2[c >> 2];
        float w = p1_w[c], bb = p1_b[c];
        a_sw += w * s1; a_sww += w * w * s2; a_swb += w * bb * s1; a_sb += bb; a_sbb += bb * bb;
        w = p2_w[c]; bb = p2_b[c];
        g_sw += w * s1; g_sww += w * w * s2; g_swb += w * bb * s1; g_sb += bb; g_sbb += bb * bb;
    }
    float v_sw = 0, v_sww = 0, v_swb = 0, v_sb = 0, v_sbb = 0;
    for (int j = tid; j < Cv_ * NH_; j += blockDim.x) {
        float s1 = Sv1[j >> 3], s2 = Sv2[j >> 3];
        float w = f1_w[j], bb = f1_b[j];
        v_sw += w * s1; v_sww += w * w * s2; v_swb += w * bb * s1; v_sb += bb; v_sbb += bb * bb;
    }
    float k_sw = 0, k_sww = 0, k_swb = 0, k_sb = 0, k_sbb = 0;
    for (int c = tid; c < Cv_; c += blockDim.x) {
        float s1 = Sv1[c], s2 = Sv2[c];
        float w = f2_w[c], bb = f2_b[c];
        k_sw += w * s1; k_sww += w * w * s2; k_swb += w * bb * s1; k_sb += bb; k_sbb += bb * bb;
    }
    a_sw = blockReduceSum(a_sw); a_sww = blockReduceSum(a_sww); a_swb = blockReduceSum(a_swb);
    a_sb = blockReduceSum(a_sb); a_sbb = blockReduceSum(a_sbb);
    g_sw = blockReduceSum(g_sw); g_sww = blockReduceSum(g_sww); g_swb = blockReduceSum(g_swb);
    g_sb = blockReduceSum(g_sb); g_sbb = blockReduceSum(g_sbb);
    v_sw = blockReduceSum(v_sw); v_sww = blockReduceSum(v_sww); v_swb = blockReduceSum(v_swb);
    v_sb = blockReduceSum(v_sb); v_sbb = blockReduceSum(v_sbb);
    k_sw = blockReduceSum(k_sw); k_sww = blockReduceSum(k_sww); k_swb = blockReduceSum(k_swb);
    k_sb = blockReduceSum(k_sb); k_sbb = blockReduceSum(k_sbb);

    if (tid == 0) {
        const float TaF = (float)(Ta_ * F_);
        const float N1 = (float)Cv_ * TaF;
        float mu1 = (a_sw + TaF * a_sb) / N1;
        float rs1 = rsqrtf((a_sww + 2.f * a_swb + TaF * a_sbb) / N1 - mu1 * mu1 + 1e-5f);
        float mu2 = (g_sw + TaF * g_sb) / N1;
        float rs2 = rsqrtf((g_sww + 2.f * g_swb + TaF * g_sbb) / N1 - mu2 * mu2 + 1e-5f);
        const float N3 = (float)(Cv_ * NH_) * (float)Tv_;
        float mu3 = (v_sw + (float)Tv_ * v_sb) / N3;
        float rs3 = rsqrtf((v_sww + 2.f * v_swb + (float)Tv_ * v_sbb) / N3 - mu3 * mu3 + 1e-5f);
        const float N4 = (float)Cv_ * (float)Tv_;
        float mu4 = (k_sw + (float)Tv_ * k_sb) / N4;
        float rs4 = rsqrtf((k_sww + 2.f * k_swb + (float)Tv_ * k_sbb) / N4 - mu4 * mu4 + 1e-5f);
        float* sc = ws + WS_SC + b * 8;
        sc[0] = mu1; sc[1] = rs1; sc[2] = mu2; sc[3] = rs2;
        sc[4] = mu3; sc[5] = rs3; sc[6] = mu4; sc[7] = rs4;
    }
}

// K3: one block (128 thr = 4 waves) per (b,c).
// WMMA f32 16x16x4 with B=ones computes the ReLU'd gate row sums (Sum over F).
__global__ __launch_bounds__(128) void k_main(
    const float* __restrict__ audio, const float* __restrict__ video,
    const float* __restrict__ p1_w, const float* __restrict__ p1_b,
    const float* __restrict__ p1_g, const float* __restrict__ p1_be,
    const float* __restrict__ p2_w, const float* __restrict__ p2_b,
    const float* __restrict__ p2_g, const float* __restrict__ p2_be,
    const float* __restrict__ f1_w, const float* __restrict__ f1_b,
    const float* __restrict__ f1_g, const float* __restrict__ f1_be,
    const float* __restrict__ f2_w, const float* __restrict__ f2_b,
    const float* __restrict__ f2_g, const float* __restrict__ f2_be,
    const float* __restrict__ ws, float* __restrict__ out) {
    __shared__ float sg[Ta_];   // Sum_f relu(gate)
    __shared__ float sa[Ta_];   // Sum_f a_val
    int blk = blockIdx.x;
    int b = blk / Cv_, c = blk % Cv_;
    int tid = threadIdx.x;

    const float* sc = ws + WS_SC + b * 8;
    float mu1 = sc[0], rs1 = sc[1], mu2 = sc[2], rs2 = sc[3];
    float mu3 = sc[4], rs3 = sc[5], mu4 = sc[6], rs4 = sc[7];

    // gate: relu(a2*x + b2)
    float a2 = p2_w[c] * rs2 * p2_g[c];
    float b2 = (p2_b[c] - mu2) * rs2 * p2_g[c] + p2_be[c];

    // --- WMMA row sums of relu(gate) over F, 16 rows per wave ---
    const float* ab = audio + (size_t)(b * Ca_ + (c >> 2)) * (Ta_ * F_);
    int wave = tid >> 5, lane = tid & 31;
    int m  = lane & 15;
    int kh = (lane >> 4) << 1;            // 0 for lanes 0-15 (K=0,1); 2 for lanes 16-31 (K=2,3)
    int r0 = wave << 4;
    const float* rowp = ab + (r0 + m) * F_ + kh;
    v8f acc = {};
    v2f bone = {1.f, 1.f};                // B = ones(4x16)
    for (int k = 0; k < F_; k += 4) {
        float x0 = rowp[k];
        float x1 = rowp[k + 1];
        v2f a;
        a[0] = fmaxf(a2 * x0 + b2, 0.f);
        a[1] = fmaxf(a2 * x1 + b2, 0.f);
        // D = A(16x4) * ones(4x16) + C  -> every column of D = row sums
        acc = __builtin_amdgcn_wmma_f32_16x16x4_f32(false, a, false, bone,
                                                    (short)0, acc, false, false);
    }
    if (lane == 0)  { for (int v = 0; v < 8; ++v) sg[r0 + v]     = acc[v]; }
    if (lane == 16) { for (int v = 0; v < 8; ++v) sg[r0 + 8 + v] = acc[v]; }

    // --- a_val F-sums from precomputed audio row sums ---
    if (tid < Ta_) {
        float r = ws[WS_ROWSUM + (b * Ca_ + (c >> 2)) * Ta_ + tid];
        sa[tid] = (p1_w[c] * r + (float)F_ * (p1_b[c] - mu1)) * rs1 * p1_g[c]
                  + (float)F_ * p1_be[c];
    }

    // --- vm (attention logit) coefficients: mean over heads folds to affine ---
    float wg = 0.f, cb = 0.f;
    for (int h = 0; h < NH_; ++h) {
        int j = c * NH_ + h;
        float w = f1_w[j], g = f1_g[j];
        wg += w * g;
        cb += (f1_b[j] - mu3) * rs3 * g + f1_be[j];
    }
    float kv = rs3 * wg * (1.f / (float)NH_);
    cb *= (1.f / (float)NH_);

    // v_key affine coefficients
    float a4 = f2_w[c] * rs4 * f2_g[c];
    float b4 = (f2_b[c] - mu4) * rs4 * f2_g[c] + f2_be[c];

    __syncthreads();

    // --- softmax over Tv=256 (2 elems/thread) + fused output ---
    size_t vbase = (size_t)(b * Cv_ + c) * Tv_;
    float v0 = video[vbase + tid], v1 = video[vbase + tid + 128];
    float vm0 = kv * v0 + cb, vm1 = kv * v1 + cb;
    float mx = blockReduceMax(fmaxf(vm0, vm1));
    float e0 = expf(vm0 - mx), e1 = expf(vm1 - mx);
    float ssum = blockReduceSum(e0 + e1);
    float inv = 1.f / ssum;
    float key0 = a4 * v0 + b4, key1 = a4 * v1 + b4;
    int t0 = tid, t1 = tid + 128;
    out[vbase + t0] = e0 * inv * sa[t0 >> 2] + key0 * sg[t0 >> 2] + v0;
    out[vbase + t1] = e1 * inv * sa[t1 >> 2] + key1 * sg[t1 >> 2] + v1;
}

extern "C" void kernel_launch(void* const* d_in, const int* in_sizes, int n_in,
                              void* d_out, int out_size, void* d_ws, size_t ws_size,
                              hipStream_t stream) {
    (void)in_sizes; (void)n_in; (void)out_size; (void)ws_size;
    const float* audio = (const float*)d_in[0];
    const float* video = (const float*)d_in[1];
    const float* p1_w  = (const float*)d_in[2];
    const float* p1_b  = (const float*)d_in[3];
    const float* p1_g  = (const float*)d_in[4];
    const float* p1_be = (const float*)d_in[5];
    const float* p2_w  = (const float*)d_in[6];
    const float* p2_b  = (const float*)d_in[7];
    const float* p2_g  = (const float*)d_in[8];
    const float* p2_be = (const float*)d_in[9];
    const float* f1_w  = (const float*)d_in[10];
    const float* f1_b  = (const float*)d_in[11];
    const float* f1_g  = (const float*)d_in[12];
    const float* f1_be = (const float*)d_in[13];
    const float* f2_w  = (const float*)d_in[14];
    const float* f2_b  = (const float*)d_in[15];
    const float* f2_g  = (const float*)d_in[16];
    const float* f2_be = (const float*)d_in[17];
    float* ws  = (float*)d_ws;
    float* out = (float*)d_out;

    k_stats  <<<B_ * Ca_ + B_ * Cv_, 256, 0, stream>>>(audio, video, ws);
    k_scalars<<<B_, 256, 0, stream>>>(p1_w, p1_b, p2_w, p2_b, f1_w, f1_b, f2_w, f2_b, ws);
    k_main   <<<B_ * Cv_, 128, 0, stream>>>(audio, video,
                                            p1_w, p1_b, p1_g, p1_be,
                                            p2_w, p2_b, p2_g, p2_be,
                                            f1_w, f1_b, f1_g, f1_be,
                                            f2_w, f2_b, f2_g, f2_be,
                                            ws, out);
}